// Neural2DOffsetMinSumDecoder_13640816132468
// MI455X (gfx1250) — compile-verified
//
#include <hip/hip_runtime.h>
#include <stdint.h>

// (3,6)-regular LDPC offset-min-sum decoder for MI455X (gfx1250, wave32).
// Working set (~20 MB) is L2-resident (192 MB L2); limiter is the E random
// gathers + E random scatters per iteration in the check pass. Adjacency is
// stored slot-major (SoA) so all 6 adjacency loads per check are fully
// coalesced (one 128B line per wave32 per load).

#define DVAR 3
#define DCHK 6

// ---------------------------------------------------------------------------
// Setup: v2c[e] = llr[edge_var[e]]  and build check -> edge adjacency in
// slot-major layout: chk_edges[slot * n_chk + c] = e.
// Slot assignment order is nondeterministic (atomicAdd), but all consumers
// are order-invariant (min / min2 / global-edge-id argmin / sign-bit XOR
// parity), so the final output is deterministic.
// ---------------------------------------------------------------------------
__global__ void ldpc_init_build(const float* __restrict__ llr,
                                const int*   __restrict__ edge_var,
                                const int*   __restrict__ edge_chk,
                                float*       __restrict__ v2c,
                                int*         __restrict__ chk_edges,
                                int*         __restrict__ cnt,
                                int E, int n_chk)
{
    int e = blockIdx.x * blockDim.x + threadIdx.x;
    if (e >= E) return;
    v2c[e] = llr[edge_var[e]];
    int c = edge_chk[e];
    int slot = atomicAdd(&cnt[c], 1);
    chk_edges[slot * n_chk + c] = e;   // SoA: slot-major
}

// ---------------------------------------------------------------------------
// Check-node pass: per check, gather 6 v2c messages, compute min1 / argmin
// (ties broken by smallest GLOBAL edge id, matching the reference's
// segment_min over e_idx) / min2, sign parity excluding self, then scatter
// c2v[e] = sprod * sgn_e * (relu(raw - beta) - alpha).
// Signs handled in bit ops; jnp.sign(0)==0 semantics via any-zero flag.
// ---------------------------------------------------------------------------
__global__ void ldpc_check_pass(const float* __restrict__ v2c,
                                float*       __restrict__ c2v,
                                const int*   __restrict__ chk_edges,
                                const float* __restrict__ beta,
                                const float* __restrict__ alpha,
                                int t, int n_chk)
{
    int c = blockIdx.x * blockDim.x + threadIdx.x;
    if (c >= n_chk) return;

    const float b = beta[t];
    const float a = alpha[t];

    // 6 fully-coalesced adjacency loads (slot-major layout).
    int e[DCHK];
#pragma unroll
    for (int i = 0; i < DCHK; ++i) e[i] = chk_edges[i * n_chk + c];

    // Prefetch the 6 scatter targets (global_prefetch_b8, gfx1250) so the
    // store lines migrate toward this WGP while the gathers + min network run.
#pragma unroll
    for (int i = 0; i < DCHK; ++i) __builtin_prefetch(&c2v[e[i]], 1, 3);

    float    m[DCHK];
    unsigned sb[DCHK];
    unsigned sall = 0u;
    int      anyzero = 0;
#pragma unroll
    for (int i = 0; i < DCHK; ++i) {
        unsigned u = __float_as_uint(v2c[e[i]]);
        sb[i] = u & 0x80000000u;
        sall ^= sb[i];
        unsigned mu = u & 0x7fffffffu;
        anyzero |= (mu == 0u);
        m[i] = __uint_as_float(mu);
    }

    float min1 = m[0];
#pragma unroll
    for (int i = 1; i < DCHK; ++i) min1 = fminf(min1, m[i]);

    // First (smallest global edge id) edge achieving min1.
    int bestI = 0;
    int bestE = 0x7fffffff;
#pragma unroll
    for (int i = 0; i < DCHK; ++i) {
        if (m[i] == min1 && e[i] < bestE) { bestE = e[i]; bestI = i; }
    }

    float min2 = __uint_as_float(0x7f800000u); // +inf
#pragma unroll
    for (int i = 0; i < DCHK; ++i) {
        if (i != bestI) min2 = fminf(min2, m[i]);
    }

#pragma unroll
    for (int i = 0; i < DCHK; ++i) {
        float raw = (i == bestI) ? min2 : min1;
        float val = fmaxf(raw - b, 0.0f) - a;
        // multiply by (+/-1) = sign-bit flip; any zero input -> c2v == 0
        float out = anyzero ? 0.0f
                            : __uint_as_float(__float_as_uint(val) ^ (sall ^ sb[i]));
        c2v[e[i]] = out;
    }
}

// ---------------------------------------------------------------------------
// Variable-node pass: edges of var v are contiguous (3v, 3v+1, 3v+2).
// v2c_new[e] = llr[v] + tot - c2v[e]    (fully contiguous b96-class traffic)
// ---------------------------------------------------------------------------
__global__ void ldpc_var_pass(const float* __restrict__ llr,
                              const float* __restrict__ c2v,
                              float*       __restrict__ v2c,
                              int n)
{
    int v = blockIdx.x * blockDim.x + threadIdx.x;
    if (v >= n) return;
    int base = v * DVAR;
    float c0 = c2v[base + 0];
    float c1 = c2v[base + 1];
    float c2 = c2v[base + 2];
    float tot = (c0 + c1) + c2;
    float L = llr[v];
    v2c[base + 0] = L + tot - c0;
    v2c[base + 1] = L + tot - c1;
    v2c[base + 2] = L + tot - c2;
}

// ---------------------------------------------------------------------------
// Final pass (replaces the last variable pass): posterior + hard decision.
// d_out = [decoded_bits int32 x N | posterior f32 x N]
// ---------------------------------------------------------------------------
__global__ void ldpc_final_pass(const float* __restrict__ llr,
                                const float* __restrict__ c2v,
                                int*         __restrict__ bits,
                                float*       __restrict__ post,
                                int n)
{
    int v = blockIdx.x * blockDim.x + threadIdx.x;
    if (v >= n) return;
    int base = v * DVAR;
    float c0 = c2v[base + 0];
    float c1 = c2v[base + 1];
    float c2 = c2v[base + 2];
    float p = llr[v] + ((c0 + c1) + c2);
    bits[v] = (p < 0.0f) ? 1 : 0;
    post[v] = p;
}

extern "C" void kernel_launch(void* const* d_in, const int* in_sizes, int n_in,
                              void* d_out, int out_size, void* d_ws, size_t ws_size,
                              hipStream_t stream) {
    const float* llr      = (const float*)d_in[0];
    const float* beta     = (const float*)d_in[1];
    const float* alpha    = (const float*)d_in[2];
    const int*   edge_var = (const int*)d_in[3];
    const int*   edge_chk = (const int*)d_in[4];
    // d_in[5] = n_checks (device scalar); derive on host from E instead.

    const int N = in_sizes[0];      // 524288 variables
    const int T = in_sizes[1];      // 10 iterations
    const int E = in_sizes[3];      // 1572864 edges
    const int n_chk = E / DCHK;     // 262144 checks

    // Workspace layout (all fully initialized each call; ws is poisoned once):
    float* v2c       = (float*)d_ws;           // E floats
    float* c2v       = v2c + E;                // E floats
    int*   chk_edges = (int*)(c2v + E);        // E ints (slot-major, 6 x n_chk)
    int*   cnt       = chk_edges + E;          // n_chk ints

    const int tpb = 256; // 8 wave32s per block

    hipMemsetAsync(cnt, 0, (size_t)n_chk * sizeof(int), stream);
    ldpc_init_build<<<(E + tpb - 1) / tpb, tpb, 0, stream>>>(
        llr, edge_var, edge_chk, v2c, chk_edges, cnt, E, n_chk);

    for (int t = 0; t < T; ++t) {
        ldpc_check_pass<<<(n_chk + tpb - 1) / tpb, tpb, 0, stream>>>(
            v2c, c2v, chk_edges, beta, alpha, t, n_chk);
        if (t + 1 < T) {
            ldpc_var_pass<<<(N + tpb - 1) / tpb, tpb, 0, stream>>>(
                llr, c2v, v2c, N);
        }
    }

    int*   bits = (int*)d_out;
    float* post = (float*)d_out + N;
    ldpc_final_pass<<<(N + tpb - 1) / tpb, tpb, 0, stream>>>(
        llr, c2v, bits, post, N);
}